// RefTorchNaive_23433341567541
// MI455X (gfx1250) — compile-verified
//
#include <hip/hip_runtime.h>
#include <cstdint>

// Problem constants (match reference)
#define B_   4
#define N_   8192
#define E_   256
#define NG   128      // group size n
#define M_   64       // N_/NG groups
#define C_   4        // channel chunks
#define CE   64       // E_/C_ features per chunk
#define ALPHA 0.5f
#define MOM   0.9f

// ---------------------------------------------------------------------------
// gfx1250 TRANS tanh if available (v_tanh_f32), else exp2/rcp expansion
// tanh(x) = 1 - 2/(exp(2x)+1), exp(2x) = exp2(2*log2(e)*x)
// ---------------------------------------------------------------------------
__device__ __forceinline__ float fast_tanh(float x) {
#if __has_builtin(__builtin_amdgcn_tanhf)
  return __builtin_amdgcn_tanhf(x);
#else
  float e = __builtin_amdgcn_exp2f(2.8853900817779268f * x);
  return 1.0f - 2.0f * __builtin_amdgcn_rcpf(e + 1.0f);
#endif
}

// ---------------------------------------------------------------------------
// Async global->LDS copy of 32 consecutive floats (8 x b128), CDNA5 ASYNCcnt
// path.  In GV mode both the LDS destination (vdst) and the global address
// (vaddr) are advanced by INST_OFFSET, so one base pair serves all 8 chunks.
// Caller must guarantee 16B alignment of both addresses.
// ---------------------------------------------------------------------------
__device__ __forceinline__ void async_tile_load_32f(const float* __restrict__ gsrc,
                                                    float* lds_dst) {
  uint32_t loff = (uint32_t)(uintptr_t)lds_dst;   // low 32 bits == LDS offset
  uint64_t ga   = (uint64_t)(uintptr_t)gsrc;
  asm volatile(
      "global_load_async_to_lds_b128 %0, %1, off\n\t"
      "global_load_async_to_lds_b128 %0, %1, off offset:16\n\t"
      "global_load_async_to_lds_b128 %0, %1, off offset:32\n\t"
      "global_load_async_to_lds_b128 %0, %1, off offset:48\n\t"
      "global_load_async_to_lds_b128 %0, %1, off offset:64\n\t"
      "global_load_async_to_lds_b128 %0, %1, off offset:80\n\t"
      "global_load_async_to_lds_b128 %0, %1, off offset:96\n\t"
      "global_load_async_to_lds_b128 %0, %1, off offset:112"
      :: "v"(loff), "v"(ga) : "memory");
}

__device__ __forceinline__ void wait_async0() {
  asm volatile("s_wait_asynccnt 0" ::: "memory");
}

// ---------------------------------------------------------------------------
// One layer of the permuted-group tanh-EMA scan.
// Block = one (b, group g, channel chunk c); 256 threads = 8 wave32.
// Thread t: row o = t>>1, feature half eb = (t&1)*32.  State in registers.
// Owner-ahead scheme: the 2 lanes owning row j+1 publish their updated row
// (and its Wj-dot t2) at the END of step j; one barrier per step.
// ---------------------------------------------------------------------------
__global__ __launch_bounds__(256) void scan_layer(
    const float* __restrict__ xin,  const float* __restrict__ xain,
    float* __restrict__ xout,       float* __restrict__ xaout,
    const float* __restrict__ W, int layer, int sshift) {

  __shared__ float Wi_sh[CE];
  __shared__ float Wj_sh[CE];
  __shared__ __align__(16) float xj_sh[2][CE];   // double-buffered row-j bcast
  __shared__ float t2_sh[2];
  __shared__ __align__(16) float stage[NG * CE]; // 32 KB async staging tile

  const int t    = threadIdx.x;
  const int o    = t >> 1;
  const int half = t & 1;
  const int eb   = half * 32;

  const int c = blockIdx.x & (C_ - 1);
  const int g = (blockIdx.x >> 2) & (M_ - 1);
  const int b = blockIdx.x >> 8;

  // Per-channel weight slices Wi[c][:], Wj[c][:] (uniform branch, 2 loads)
  if (t < CE) {
    const float* Wl = W + layer * 2 * E_ + c * CE;
    Wi_sh[t] = Wl[t];
    Wj_sh[t] = Wl[E_ + t];
  }

  // perm: global row for position o of group g = ((g + o*s) % m)*n + o
  const int    row  = ((g + o * sshift) & (M_ - 1)) * NG + o;
  const size_t base = ((size_t)(b * N_ + row)) * E_ + c * CE + eb;

  float xi[32], xa[32], wi[32];

  // ---- x tile: async global->LDS (ASYNCcnt); xa tile: direct loads (LOADcnt)
  // The two streams overlap; async waits don't block the vector loads.
  float* myStage = &stage[o * CE + eb];
  async_tile_load_32f(xin + base, myStage);

  const float4* pa = (const float4*)(xain + base);
#pragma unroll
  for (int q = 0; q < 8; ++q) {
    float4 v = pa[q];
    xa[4 * q] = v.x; xa[4 * q + 1] = v.y; xa[4 * q + 2] = v.z; xa[4 * q + 3] = v.w;
  }

  wait_async0();
  __syncthreads();                          // stage + weights visible

  const float4* sp = (const float4*)myStage;
#pragma unroll
  for (int q = 0; q < 8; ++q) {
    float4 v = sp[q];
    xi[4 * q] = v.x; xi[4 * q + 1] = v.y; xi[4 * q + 2] = v.z; xi[4 * q + 3] = v.w;
  }

#pragma unroll
  for (int k = 0; k < 32; ++k) wi[k] = Wi_sh[eb + k];

  // t1[o] = xi[o] . Wi  (half-row lanes combine via wave32 xor-shuffle)
  float p = 0.f;
#pragma unroll
  for (int k = 0; k < 32; ++k) p += xi[k] * wi[k];
  float t1 = p + __shfl_xor(p, 1, 32);

  // Prologue: owners of row 0 publish snapshot + t2 for step 0
  if (o == 0) {
    float4* wp = (float4*)&xj_sh[0][eb];
    float q2 = 0.f;
#pragma unroll
    for (int q = 0; q < 8; ++q) {
      wp[q] = make_float4(xi[4 * q], xi[4 * q + 1], xi[4 * q + 2], xi[4 * q + 3]);
      q2 += xi[4 * q]     * Wj_sh[eb + 4 * q]
          + xi[4 * q + 1] * Wj_sh[eb + 4 * q + 1]
          + xi[4 * q + 2] * Wj_sh[eb + 4 * q + 2]
          + xi[4 * q + 3] * Wj_sh[eb + 4 * q + 3];
    }
    q2 += __shfl_xor(q2, 1, 32);
    if (half == 0) t2_sh[0] = q2;
  }
  __syncthreads();

  // ---- sequential scan over j = 0..127 ----
  for (int j = 0; j < NG; ++j) {
    const int buf = j & 1;
    const float  Wij = t1 + t2_sh[buf];           // per-row scalar
    const float4* xjp = (const float4*)&xj_sh[buf][eb];

    float pn = 0.f;
#pragma unroll
    for (int q = 0; q < 8; ++q) {
      const float4 xj4 = xjp[q];                  // ds_load_b128
      const float xje[4] = {xj4.x, xj4.y, xj4.z, xj4.w};
#pragma unroll
      for (int r = 0; r < 4; ++r) {
        const int k = 4 * q + r;
        const float T = ALPHA * xi[k] + (1.0f - ALPHA) * Wij * xje[r];
        const float F = fast_tanh(T);
        xa[k] = MOM * xa[k] + (1.0f - MOM) * F;     // ya
        xi[k] = MOM * xi[k] + (1.0f - MOM) * xa[k]; // yi
        pn += xi[k] * wi[k];                        // fold next-step t1 dot in
      }
    }
    t1 = pn + __shfl_xor(pn, 1, 32);

    // Owner-ahead: lanes owning row j+1 publish updated row + t2 for step j+1.
    // (o ranges 0..127, so at j==127 no thread matches and this vanishes.)
    if (o == j + 1) {
      const int nb = buf ^ 1;
      float4* wp = (float4*)&xj_sh[nb][eb];
      float q2 = 0.f;
#pragma unroll
      for (int q = 0; q < 8; ++q) {
        wp[q] = make_float4(xi[4 * q], xi[4 * q + 1], xi[4 * q + 2], xi[4 * q + 3]);
        q2 += xi[4 * q]     * Wj_sh[eb + 4 * q]
            + xi[4 * q + 1] * Wj_sh[eb + 4 * q + 1]
            + xi[4 * q + 2] * Wj_sh[eb + 4 * q + 2]
            + xi[4 * q + 3] * Wj_sh[eb + 4 * q + 3];
      }
      q2 += __shfl_xor(q2, 1, 32);
      if (half == 0) t2_sh[nb] = q2;
    }
    __syncthreads();                              // one barrier per step
  }

  // ---- scatter-back == write to the exact gathered locations ----
  float4* ox  = (float4*)(xout  + base);
  float4* oxa = (float4*)(xaout + base);
#pragma unroll
  for (int q = 0; q < 8; ++q) {
    ox[q]  = make_float4(xi[4 * q], xi[4 * q + 1], xi[4 * q + 2], xi[4 * q + 3]);
    oxa[q] = make_float4(xa[4 * q], xa[4 * q + 1], xa[4 * q + 2], xa[4 * q + 3]);
  }
}

// ---------------------------------------------------------------------------
extern "C" void kernel_launch(void* const* d_in, const int* in_sizes, int n_in,
                              void* d_out, int out_size, void* d_ws, size_t ws_size,
                              hipStream_t stream) {
  (void)in_sizes; (void)n_in; (void)out_size; (void)d_ws; (void)ws_size;

  const float* x  = (const float*)d_in[0];
  const float* xa = (const float*)d_in[1];
  const float* W  = (const float*)d_in[2];

  float* xo  = (float*)d_out;                       // x output half
  float* xao = xo + (size_t)B_ * N_ * E_;           // xa output half

  dim3 grid(B_ * M_ * C_);   // 1024 blocks
  dim3 block(256);

  // Layer 0: identity perm (s=0), d_in -> d_out
  scan_layer<<<grid, block, 0, stream>>>(x, xa, xo, xao, W, 0, 0);
  // Layer 1: s = (1<<0) % m = 1, in place on d_out (element ownership is a
  // bijection per block: every element is read and written by the same block)
  scan_layer<<<grid, block, 0, stream>>>(xo, xao, xo, xao, W, 1, 1);
}